// VectorQuantizer_54056458387889
// MI455X (gfx1250) — compile-verified
//
#include <hip/hip_runtime.h>

typedef __attribute__((ext_vector_type(2))) float v2f;
typedef __attribute__((ext_vector_type(8))) float v8f;

#define K_CODES 1024
#define D_DIM   256
#define N_ROWS  32768
#define TILE_M  64          // 4 row-tiles of 16 -> 4x less L2 codebook traffic
#define RT      4           // row tiles per block
#define LDS_STRIDE 260      // 260 % 64 banks = 4 -> conflict-free 8B row accesses

// ---------------------------------------------------------------------------
// Kernel 1: zero histogram + SSE accumulator, precompute |e_k|^2 in f32
// ---------------------------------------------------------------------------
__global__ __launch_bounds__(256)
void vq_init(float* __restrict__ e_norm, unsigned int* __restrict__ counts,
             float* __restrict__ sse, const float* __restrict__ weight) {
    int t = blockIdx.x * blockDim.x + threadIdx.x;   // 1024 threads total
    if (t < K_CODES) {
        counts[t] = 0u;
        const float4* wrow = (const float4*)(weight + (size_t)t * D_DIM);
        float s = 0.f;
        for (int i = 0; i < D_DIM / 4; ++i) {
            float4 w = wrow[i];
            s += w.x * w.x + w.y * w.y + w.z * w.z + w.w * w.w;
        }
        e_norm[t] = s;
    }
    if (t == 0) *sse = 0.f;
}

// ---------------------------------------------------------------------------
// Kernel 2: 64-row tile per block. Each wave: 4 col-tile pairs (128 codes),
// 4x2 register block -> one B-fragment pair feeds 8 independent
// v_wmma_f32_16x16x4_f32 accumulator chains (B bytes/WMMA cut 4x, deep
// instruction-level overlap). Then argmin + all outputs.
// ---------------------------------------------------------------------------
__global__ __launch_bounds__(256)
void vq_main(const float* __restrict__ data,
             const float* __restrict__ weight,
             const float* __restrict__ e_norm,
             unsigned int* __restrict__ counts,
             float* __restrict__ sse,
             float* __restrict__ out_q,
             float* __restrict__ out_enc) {
    __shared__ float tile[TILE_M * LDS_STRIDE];
    __shared__ float red_d[TILE_M][8];
    __shared__ int   red_i[TILE_M][8];
    __shared__ int   final_idx[TILE_M];

    const int tid  = threadIdx.x;
    const int wave = tid >> 5;
    const int lane = tid & 31;
    const long row0 = (long)blockIdx.x * TILE_M;

    // Stage 64x256 f32 data tile into LDS (padded stride), float4 loads.
    {
        const float4* src = (const float4*)(data + row0 * D_DIM);
        for (int i = tid; i < TILE_M * D_DIM / 4; i += 256) {
            int r  = i >> 6;        // 64 float4 per row
            int c4 = i & 63;
            float4 v = src[i];
            *(float4*)(&tile[r * LDS_STRIDE + c4 * 4]) = v;
        }
    }
    __syncthreads();

    const int m  = lane & 15;            // A-matrix row / B-matrix column
    const int kb = (lane >> 4) * 2;      // K sub-offset per 16-lane half

    float bestD[RT][8];
    int   bestI[RT][8];
#pragma unroll
    for (int rt = 0; rt < RT; ++rt)
#pragma unroll
        for (int v = 0; v < 8; ++v) { bestD[rt][v] = 3.4e38f; bestI[rt][v] = 0; }

    // Each wave scans its 8 column tiles as 4 pairs; 8 waves cover K=1024.
    for (int p = 0; p < 4; ++p) {
        const int ct0  = wave * 8 + 2 * p;
        const int col0 = ct0 * 16 + m;
        const int col1 = col0 + 16;
        const float* __restrict__ wrow0 = weight + (size_t)col0 * D_DIM;
        const float* __restrict__ wrow1 = weight + (size_t)col1 * D_DIM;

        v8f acc[RT][2];
#pragma unroll
        for (int rt = 0; rt < RT; ++rt) { acc[rt][0] = (v8f){}; acc[rt][1] = (v8f){}; }

        for (int dstep = 0; dstep < 64; ++dstep) {
            const int d = dstep * 4 + kb;
            v2f b0, b1;
            b0.x = wrow0[d];
            b0.y = wrow0[d + 1];
            b1.x = wrow1[d];
            b1.y = wrow1[d + 1];
#pragma unroll
            for (int rt = 0; rt < RT; ++rt) {
                v2f a;
                a.x = tile[(rt * 16 + m) * LDS_STRIDE + d];
                a.y = tile[(rt * 16 + m) * LDS_STRIDE + d + 1];
                acc[rt][0] = __builtin_amdgcn_wmma_f32_16x16x4_f32(
                    false, a, false, b0, (short)0, acc[rt][0], false, false);
                acc[rt][1] = __builtin_amdgcn_wmma_f32_16x16x4_f32(
                    false, a, false, b1, (short)0, acc[rt][1], false, false);
            }
        }

        const float en0 = e_norm[col0];
        const float en1 = e_norm[col1];
#pragma unroll
        for (int rt = 0; rt < RT; ++rt)
#pragma unroll
            for (int v = 0; v < 8; ++v) {
                float dist0 = en0 - 2.0f * acc[rt][0][v];
                float dist1 = en1 - 2.0f * acc[rt][1][v];
                if (dist0 < bestD[rt][v]) { bestD[rt][v] = dist0; bestI[rt][v] = col0; }
                if (dist1 < bestD[rt][v]) { bestD[rt][v] = dist1; bestI[rt][v] = col1; }
            }
    }

    // C/D layout: vgpr v, lanes 0-15 -> row rt*16+v; lanes 16-31 -> rt*16+v+8.
    // Min-reduce within each 16-lane half (argmin over this wave's columns).
#pragma unroll
    for (int rt = 0; rt < RT; ++rt)
#pragma unroll
        for (int v = 0; v < 8; ++v) {
            float d = bestD[rt][v];
            int   idx = bestI[rt][v];
#pragma unroll
            for (int off = 8; off > 0; off >>= 1) {
                float od = __shfl_xor(d,   off, 16);
                int   oi = __shfl_xor(idx, off, 16);
                if (od < d || (od == d && oi < idx)) { d = od; idx = oi; }
            }
            if (m == 0) {
                int r = rt * 16 + v + ((lane >> 4) << 3);
                red_d[r][wave] = d;
                red_i[r][wave] = idx;
            }
        }
    __syncthreads();

    // Cross-wave argmin per row, final code index; bump histogram.
    if (tid < TILE_M) {
        float d = red_d[tid][0];
        int idx  = red_i[tid][0];
        for (int w = 1; w < 8; ++w) {
            float od = red_d[tid][w];
            int   oi = red_i[tid][w];
            if (od < d || (od == d && oi < idx)) { d = od; idx = oi; }
        }
        final_idx[tid] = idx;
        atomicAdd(&counts[idx], 1u);
    }
    __syncthreads();

    // quantized = gather(codebook); accumulate SSE for loss (coalesced).
    float local_sse = 0.f;
    for (int i = tid; i < TILE_M * D_DIM; i += 256) {
        int r = i >> 8, c = i & 255;
        float q = weight[(size_t)final_idx[r] * D_DIM + c];
        float x = tile[r * LDS_STRIDE + c];
        float diff = q - x;
        local_sse += diff * diff;
        out_q[(row0 + r) * D_DIM + c] = q;
    }
#pragma unroll
    for (int off = 16; off > 0; off >>= 1)
        local_sse += __shfl_xor(local_sse, off, 32);
    if (lane == 0) atomicAdd(sse, local_sse);

    // one-hot encodings: 64 rows x 1024, float4 stores (dominant BW term).
    float4* enc = (float4*)(out_enc + row0 * K_CODES);
    for (int i = tid; i < TILE_M * K_CODES / 4; i += 256) {
        int r = i >> 8;              // 256 float4 per row
        int c = (i & 255) * 4;
        int fi = final_idx[r];
        float4 v = {0.f, 0.f, 0.f, 0.f};
        if (fi >= c && fi < c + 4) ((float*)&v)[fi - c] = 1.0f;
        enc[i] = v;
    }
}

// ---------------------------------------------------------------------------
// Kernel 3: scalars — loss = 1.25 * MSE ; perplexity = exp(-sum p log(p+eps))
// ---------------------------------------------------------------------------
__global__ __launch_bounds__(1024)
void vq_final(const unsigned int* __restrict__ counts,
              const float* __restrict__ sse,
              float* __restrict__ out_scalars) {
    __shared__ float partial[32];
    int tid = threadIdx.x;                       // 1024 threads
    float p = (float)counts[tid] * (1.0f / (float)N_ROWS);
    float term = p * logf(p + 1e-10f);
#pragma unroll
    for (int off = 16; off > 0; off >>= 1)
        term += __shfl_xor(term, off, 32);
    if ((tid & 31) == 0) partial[tid >> 5] = term;
    __syncthreads();
    if (tid < 32) {
        float s = partial[tid];
#pragma unroll
        for (int off = 16; off > 0; off >>= 1)
            s += __shfl_xor(s, off, 32);
        if (tid == 0) {
            out_scalars[0] = 1.25f * (*sse) / (float)((long)N_ROWS * D_DIM);
            out_scalars[1] = expf(-s);
        }
    }
}

// ---------------------------------------------------------------------------
extern "C" void kernel_launch(void* const* d_in, const int* in_sizes, int n_in,
                              void* d_out, int out_size, void* d_ws, size_t ws_size,
                              hipStream_t stream) {
    const float* data   = (const float*)d_in[0];
    const float* weight = (const float*)d_in[1];

    float* out        = (float*)d_out;
    float* out_q      = out;                                        // 32768*256
    float* out_enc    = out + (size_t)N_ROWS * D_DIM;               // 32768*1024
    float* out_scal   = out + (size_t)N_ROWS * D_DIM
                            + (size_t)N_ROWS * K_CODES;             // loss, perplexity

    float*        e_norm = (float*)d_ws;                            // 1024 f32
    unsigned int* counts = (unsigned int*)((char*)d_ws + 4096);     // 1024 u32
    float*        sse    = (float*)((char*)d_ws + 8192);            // 1 f32

    vq_init <<<4, 256, 0, stream>>>(e_norm, counts, sse, weight);
    vq_main <<<N_ROWS / TILE_M, 256, 0, stream>>>(data, weight, e_norm, counts,
                                                  sse, out_q, out_enc);
    vq_final<<<1, 1024, 0, stream>>>(counts, sse, out_scal);
}